// SSIM_5600637354725
// MI455X (gfx1250) — compile-verified
//
#include <hip/hip_runtime.h>
#include <math.h>

// SSIM (11^3 separable Gaussian, replicate padding) fused single-pass kernel.
// W/H blur passes run on the fp32 matrix pipe via V_WMMA_F32_16X16X4_F32
// (banded Gaussian matmul), D blur + SSIM formula on VALU, deterministic
// two-stage reduction (no float atomics).

typedef __attribute__((ext_vector_type(2))) float v2f;
typedef __attribute__((ext_vector_type(8))) float v8f;

#define DD 64
#define HH 512
#define WW 512
#define PADR 5
#define TH 16
#define TW 16
// padded tile: 26 rows (h) x 28 cols (w); cols 26..27 are K-padding (zeros in band)

__device__ __forceinline__ int iclamp(int v, int lo, int hi) {
  return v < lo ? lo : (v > hi ? hi : v);
}

// unnormalized gaussian tap, center 5, sigma=1.5 -> 2*sigma^2 = 4.5
__device__ __forceinline__ float gauss_raw(int j) {
  float c = (float)(j - 5);
  return expf(-(c * c) * (1.0f / 4.5f));
}

__global__ __launch_bounds__(320) void ssim_fused(
    const float* __restrict__ X, const float* __restrict__ Y,
    float* __restrict__ partial) {
  __shared__ float xt[26][28];
  __shared__ float yt[26][28];
  __shared__ float t1[5][28][16];      // W-blurred planes, k-major for B fragments
  __shared__ float ring[11][5][256];   // 2D-blurred plane ring over D

  const int tid  = threadIdx.x;
  // wave index is uniform per wave: force it into an SGPR so that all
  // field/row-group dispatch below compiles to scalar branches (EXEC untouched
  // around the WMMA chains).
  const int swave = __builtin_amdgcn_readfirstlane(tid >> 5);
  const int lane = tid & 31;
  const int hi   = lane >> 4;   // lane half (K sub-block select)
  const int li   = lane & 15;   // M / N index within half

  // --- gaussian taps (normalized) ---
  float norm = 0.f;
#pragma unroll
  for (int j = 0; j < 11; ++j) norm += gauss_raw(j);
  const float invn = 1.0f / norm;
  float g[11];
#pragma unroll
  for (int j = 0; j < 11; ++j) g[j] = gauss_raw(j) * invn;

  // --- banded gaussian matrix fragments: G[k][i] = g[k - i], k in 4-blocks ---
  // Identical VGPR layout serves as B (W pass: B[k][n]=g[k-n]) and as A
  // (H pass: A[m][k]=g[k-m]) because both split K across lane halves with
  // the other index = li.
  v2f gb[7];
#pragma unroll
  for (int kk = 0; kk < 7; ++kk) {
    int k0 = 4 * kk + 2 * hi;
    int d0 = k0 - li, d1 = k0 + 1 - li;
    gb[kk].x = (d0 >= 0 && d0 <= 10) ? gauss_raw(d0) * invn : 0.0f;
    gb[kk].y = (d1 >= 0 && d1 <= 10) ? gauss_raw(d1) * invn : 0.0f;
  }

  const int nb = blockIdx.z;           // batch index
  const int h0 = blockIdx.y * TH;
  const int w0 = blockIdx.x * TW;

  // zero the K-padding rows (26,27) of t1 once; never written afterwards
  for (int i = tid; i < 5 * 2 * 16; i += 320) {
    int f = i / 32, r = (i >> 4) & 1, c = i & 15;
    t1[f][26 + r][c] = 0.f;
  }

  // W-pass per-wave scalar roles: field f = wave>>1, row-group rg = wave&1
  const int wf  = swave >> 1;
  const int wrg = swave & 1;
  const int arow = li + 10 * wrg;      // padded h row 0..25 for A fragments

  float ssim_acc = 0.f;
  int slot = 0;                        // pd % 11, tracked incrementally

  for (int pd = 0; pd < DD + 2 * PADR; ++pd) {
    const int gd      = iclamp(pd - PADR, 0, DD - 1);
    const int gd_next = iclamp(pd + 1 - PADR, 0, DD - 1);
    const size_t pbase  = ((size_t)nb * DD + gd) * (size_t)(HH * WW);
    const size_t pbase2 = ((size_t)nb * DD + gd_next) * (size_t)(HH * WW);

    __syncthreads();  // previous plane's readers done with xt/yt
    for (int i = tid; i < 26 * 28; i += 320) {
      int lh = i / 28, lw = i - lh * 28;
      int gh = iclamp(h0 - PADR + lh, 0, HH - 1);
      int gw = iclamp(w0 - PADR + lw, 0, WW - 1);
      size_t off = (size_t)gh * WW + gw;
      xt[lh][lw] = X[pbase + off];
      yt[lh][lw] = Y[pbase + off];
      // warm L2 for the next D-plane (global_prefetch_b8)
      __builtin_prefetch(X + pbase2 + off, 0, 1);
      __builtin_prefetch(Y + pbase2 + off, 0, 1);
    }
    __syncthreads();

    // ---- W pass: out1[row][n] = sum_k F[row][k] * g[k-n], fp32 WMMA ----
    // All 10 waves participate (blockDim == 320).
    {
      // Load all K fragments first (contiguous b64 pairs), then build the
      // field-specific A fragments with ONE scalar branch, then run the
      // 7-step WMMA chain back-to-back.
      v2f xv[7], yv[7];
#pragma unroll
      for (int kk = 0; kk < 7; ++kk) {
        const int k0 = 4 * kk + 2 * hi;
        xv[kk].x = xt[arow][k0]; xv[kk].y = xt[arow][k0 + 1];
        yv[kk].x = yt[arow][k0]; yv[kk].y = yt[arow][k0 + 1];
      }
      v2f a[7];
      if (wf == 0) {
#pragma unroll
        for (int kk = 0; kk < 7; ++kk) a[kk] = xv[kk];
      } else if (wf == 1) {
#pragma unroll
        for (int kk = 0; kk < 7; ++kk) a[kk] = yv[kk];
      } else if (wf == 2) {
#pragma unroll
        for (int kk = 0; kk < 7; ++kk) { a[kk].x = xv[kk].x * xv[kk].x; a[kk].y = xv[kk].y * xv[kk].y; }
      } else if (wf == 3) {
#pragma unroll
        for (int kk = 0; kk < 7; ++kk) { a[kk].x = yv[kk].x * yv[kk].x; a[kk].y = yv[kk].y * yv[kk].y; }
      } else {
#pragma unroll
        for (int kk = 0; kk < 7; ++kk) { a[kk].x = xv[kk].x * yv[kk].x; a[kk].y = xv[kk].y * yv[kk].y; }
      }
      v8f acc = {};
#pragma unroll
      for (int kk = 0; kk < 7; ++kk) {
        acc = __builtin_amdgcn_wmma_f32_16x16x4_f32(
            false, a[kk], false, gb[kk], (short)0, acc, false, false);
      }
      // Unconditional stores: row-groups overlap on t1 rows 10..15, but both
      // groups compute those rows from identical inputs with the identical
      // instruction sequence, so the overlapping writes carry bitwise-equal
      // values -> deterministic regardless of order. No EXEC masking needed.
#pragma unroll
      for (int r = 0; r < 8; ++r) {
        const int orow = r + 8 * hi + 10 * wrg;  // rg0: 0..15, rg1: 10..25
        t1[wf][orow][li] = acc[r];
      }
    }
    __syncthreads();

    // ---- H pass: out2[m][n] = sum_k g[k-m] * t1[k][n], fp32 WMMA ----
    if (swave < 5) {
      const int f = swave;
      v2f b[7];
#pragma unroll
      for (int kk = 0; kk < 7; ++kk) {
        const int k0 = 4 * kk + 2 * hi;
        b[kk].x = t1[f][k0][li];
        b[kk].y = t1[f][k0 + 1][li];
      }
      v8f acc = {};
#pragma unroll
      for (int kk = 0; kk < 7; ++kk) {
        acc = __builtin_amdgcn_wmma_f32_16x16x4_f32(
            false, gb[kk], false, b[kk], (short)0, acc, false, false);
      }
#pragma unroll
      for (int r = 0; r < 8; ++r) {
        ring[slot][f][(r + 8 * hi) * 16 + li] = acc[r];
      }
    }
    __syncthreads();

    // ---- D pass + SSIM (element-wise over ring planes) ----
    if (pd >= 10 && swave < 8) {     // threads 0..255, one voxel each
      // base ring slot for output plane d = pd-10 is (slot+1) mod 11;
      // track slots with add/wrap instead of integer division.
      int sl = slot + 1; if (sl >= 11) sl -= 11;
      float mx = 0.f, my = 0.f, mxx = 0.f, myy = 0.f, mxy = 0.f;
#pragma unroll
      for (int j = 0; j < 11; ++j) {
        const float w = g[j];
        mx  += w * ring[sl][0][tid];
        my  += w * ring[sl][1][tid];
        mxx += w * ring[sl][2][tid];
        myy += w * ring[sl][3][tid];
        mxy += w * ring[sl][4][tid];
        ++sl; if (sl >= 11) sl -= 11;
      }
      const float mx2 = mx * mx, my2 = my * my, mu_xy = mx * my;
      float sx = mxx - mx2; sx = sx > 0.f ? sx : 0.f;
      float sy = myy - my2; sy = sy > 0.f ? sy : 0.f;
      const float sxy = mxy - mu_xy;
      const float c1 = 1.0e-4f, c2 = 9.0e-4f;
      ssim_acc += ((2.0f * mu_xy + c1) * (2.0f * sxy + c2)) /
                  ((mx2 + my2 + c1) * (sx + sy + c2));
    }

    ++slot; if (slot >= 11) slot -= 11;
  }

  // ---- block reduction (reuse xt storage) ----
  __syncthreads();
  float* red = &xt[0][0];
  if (swave < 8) red[tid] = ssim_acc;
  __syncthreads();
  for (int s = 128; s > 0; s >>= 1) {
    if (tid < s) red[tid] += red[tid + s];
    __syncthreads();
  }
  if (tid == 0) {
    const int bid = (blockIdx.z * gridDim.y + blockIdx.y) * gridDim.x + blockIdx.x;
    partial[bid] = red[0];
  }
}

__global__ __launch_bounds__(256) void reduce_partials(
    const float* __restrict__ partial, int nPartial, float* __restrict__ out) {
  __shared__ float red[256];
  float s = 0.f;
  for (int i = threadIdx.x; i < nPartial; i += 256) s += partial[i];
  red[threadIdx.x] = s;
  __syncthreads();
  for (int k = 128; k > 0; k >>= 1) {
    if (threadIdx.x < k) red[threadIdx.x] += red[threadIdx.x + k];
    __syncthreads();
  }
  if (threadIdx.x == 0) out[0] = red[0] * (1.0f / 33554432.0f);  // / (2*64*512*512)
}

extern "C" void kernel_launch(void* const* d_in, const int* in_sizes, int n_in,
                              void* d_out, int out_size, void* d_ws, size_t ws_size,
                              hipStream_t stream) {
  const float* X = (const float*)d_in[0];
  const float* Y = (const float*)d_in[1];
  float* partial = (float*)d_ws;  // 2048 floats

  dim3 grid(WW / TW, HH / TH, 2);
  ssim_fused<<<grid, 320, 0, stream>>>(X, Y, partial);
  reduce_partials<<<1, 256, 0, stream>>>(partial, (WW / TW) * (HH / TH) * 2,
                                         (float*)d_out);
}